// RefBasedDeepMetric_39651138077346
// MI455X (gfx1250) — compile-verified
//
#include <hip/hip_runtime.h>

// CDNA5 (gfx1250) wave32 WMMA f32 path: V_WMMA_F32_16X16X4_F32
// Data movement: GLOBAL_LOAD_ASYNC_TO_LDS_B128 (ASYNCcnt), double-buffered LDS.
typedef __attribute__((ext_vector_type(2))) float v2f;
typedef __attribute__((ext_vector_type(8))) float v8f;

#define NROW 4096   // B == R
#define DIN  1024
#define DOUT 1024

// Block tile: 128 (M) x 64 (N), K-stage 16. 8 waves of 32 -> each wave a 32x32 subtile.
#define BM 128
#define BN 64
#define KT 16
#define NSTAGE (DIN / KT)   // 64

// workspace layout (floats): [0,1024) colsum(refs) S, [1024,2048) mean_ref, [2048] sumsq accum
#define WS_S        0
#define WS_MEANREF  1024
#define WS_SUMSQ    2048
#define WS_NZERO    2049

// ---------------- CDNA5 async global->LDS helpers ----------------

__device__ __forceinline__ void async_ld_b128(unsigned lds_off, const void* gaddr) {
  // VDST = per-lane LDS byte address, VADDR = per-lane 64-bit global address (GV mode)
  asm volatile("global_load_async_to_lds_b128 %0, %1, off"
               :: "v"(lds_off), "v"(gaddr) : "memory");
}
__device__ __forceinline__ void wait_async_le3() {
  asm volatile("s_wait_asynccnt 0x3" ::: "memory");
}
__device__ __forceinline__ void wait_async_0() {
  asm volatile("s_wait_asynccnt 0x0" ::: "memory");
}

// ---------------- small reduction kernels ----------------

__global__ __launch_bounds__(256) void zero_ws_kernel(float* __restrict__ ws) {
  int i = blockIdx.x * blockDim.x + threadIdx.x;
  if (i < WS_NZERO) ws[i] = 0.f;
}

// Partial column sums of refs: grid (DIN/256, 32), each block covers 128 rows.
__global__ __launch_bounds__(256) void colsum_kernel(
    const float* __restrict__ refs, float* __restrict__ ws) {
  int d  = blockIdx.x * blockDim.x + threadIdx.x;   // column, coalesced
  int r0 = blockIdx.y * (NROW / 32);
  float s = 0.f;
  for (int r = r0; r < r0 + NROW / 32; ++r) s += refs[(size_t)r * DIN + d];
  atomicAdd(ws + WS_S + d, s);
}

// Partial GEMV: ws[MEANREF+d] += sum_{k in slice} S[k] * W[k][d]; grid (DOUT/256, 8)
__global__ __launch_bounds__(256) void meanref_partial_kernel(
    const float* __restrict__ W, float* __restrict__ ws) {
  int d  = blockIdx.x * blockDim.x + threadIdx.x;
  int k0 = blockIdx.y * (DIN / 8);
  float s = 0.f;
  for (int k = k0; k < k0 + DIN / 8; ++k) s += ws[WS_S + k] * W[(size_t)k * DOUT + d];
  atomicAdd(ws + WS_MEANREF + d, s);
}

__global__ __launch_bounds__(256) void meanref_finalize_kernel(
    const float* __restrict__ bias, float* __restrict__ ws) {
  int d = blockIdx.x * blockDim.x + threadIdx.x;
  ws[WS_MEANREF + d] = ws[WS_MEANREF + d] * (1.0f / (float)NROW) + bias[d];
}

__global__ __launch_bounds__(256) void init_out_kernel(
    const float* __restrict__ ws, float* __restrict__ out) {
  int i = blockIdx.x * blockDim.x + threadIdx.x;
  out[i] = ws[WS_SUMSQ] * (1.0f / (float)NROW);    // seed with mean_sq_ref
}

// ---------------- fused WMMA GEMM + reduction ----------------
// MODE 0: accum = scalar; add sum over tile of (A@W+b)^2          (refs pass)
// MODE 1: accum = out rows; add per-row sum of v^2 - 2*v*mean_ref (x pass)

template <int MODE>
__global__ __launch_bounds__(256) void wmma_gemm_kernel(
    const float* __restrict__ A,        // NROW x DIN, row major
    const float* __restrict__ W,        // DIN x DOUT, row major
    const float* __restrict__ bias,     // DOUT
    const float* __restrict__ meanref,  // DOUT (MODE 1 only)
    float* __restrict__ accum) {
  __shared__ float As[2][BM][KT];   // 16 KB (double buffered)
  __shared__ float Bs[2][KT][BN];   //  8 KB

  const int tid  = threadIdx.x;
  const int lane = tid & 31;
  const int l15  = lane & 15;
  const int h    = lane >> 4;      // half-wave select (K split per ISA layout)
  const int wid  = tid >> 5;       // 8 waves
  const int wm   = wid >> 1;       // 0..3 : wave row in block tile
  const int wn   = wid & 1;        // 0..1 : wave col in block tile

  const int rowBase = blockIdx.x * BM;
  const int colBase = blockIdx.y * BN;

  // ---- per-thread staging geometry (constant across stages) ----
  // A tile: 128x16 f32 = 512 float4, 2 per thread; B tile: 16x64 f32 = 256 float4, 1 per thread
  const int fA0 = tid * 2, fA1 = tid * 2 + 1;
  const int rA0 = fA0 >> 2, cA0 = (fA0 & 3) * 4;
  const int rA1 = fA1 >> 2, cA1 = (fA1 & 3) * 4;
  const int rB  = tid >> 4, cB  = (tid & 15) * 4;

  unsigned ldsA0[2], ldsA1[2], ldsB[2];
#pragma unroll
  for (int b = 0; b < 2; ++b) {
    ldsA0[b] = (unsigned)(uintptr_t)&As[b][rA0][cA0];
    ldsA1[b] = (unsigned)(uintptr_t)&As[b][rA1][cA1];
    ldsB[b]  = (unsigned)(uintptr_t)&Bs[b][rB][cB];
  }
  const float* gA0 = A + (size_t)(rowBase + rA0) * DIN + cA0;
  const float* gA1 = A + (size_t)(rowBase + rA1) * DIN + cA1;
  const float* gB  = W + (size_t)rB * DOUT + colBase + cB;

  v8f c[2][2] = {};                // 2x2 fragments of 16x16 f32

  // prologue: stage 0 in flight
  async_ld_b128(ldsA0[0], gA0);
  async_ld_b128(ldsA1[0], gA1);
  async_ld_b128(ldsB[0],  gB);

  for (int s = 0; s < NSTAGE; ++s) {
    if (s + 1 < NSTAGE) {
      const int k0n = (s + 1) * KT;
      const int nb  = (s + 1) & 1;
      async_ld_b128(ldsA0[nb], gA0 + k0n);
      async_ld_b128(ldsA1[nb], gA1 + k0n);
      async_ld_b128(ldsB[nb],  gB + (size_t)k0n * DOUT);
      wait_async_le3();            // retire stage s's 3 copies (in-order completion)
    } else {
      wait_async_0();
    }
    __syncthreads();               // all waves' stage-s data visible in LDS

    const float (*AsS)[KT] = As[s & 1];
    const float (*BsS)[BN] = Bs[s & 1];

#pragma unroll
    for (int kk = 0; kk < KT; kk += 4) {
      // A frag 16x4: lanes 0-15 -> K=kk,kk+1 ; lanes 16-31 -> K=kk+2,kk+3
      v2f a[2], b[2];
#pragma unroll
      for (int mt = 0; mt < 2; ++mt) {
        int row = wm * 32 + mt * 16 + l15;
        a[mt].x = AsS[row][kk + 2 * h];
        a[mt].y = AsS[row][kk + 2 * h + 1];
      }
      // B frag 4x16: VGPR0 holds rows K=kk / kk+2, VGPR1 rows K=kk+1 / kk+3
#pragma unroll
      for (int nt = 0; nt < 2; ++nt) {
        int col = wn * 32 + nt * 16 + l15;
        b[nt].x = BsS[kk + 2 * h][col];
        b[nt].y = BsS[kk + 2 * h + 1][col];
      }
#pragma unroll
      for (int mt = 0; mt < 2; ++mt)
#pragma unroll
        for (int nt = 0; nt < 2; ++nt)
          c[mt][nt] = __builtin_amdgcn_wmma_f32_16x16x4_f32(
              false, a[mt], false, b[nt], (short)0, c[mt][nt], false, false);
    }
    __syncthreads();               // done reading buf before it's refilled
  }

  // ---- epilogue ----
  // C/D layout: VGPR i, lanes 0-15: (M=i, N=l15); lanes 16-31: (M=i+8, N=l15)
  if (MODE == 0) {
    float s = 0.f;
#pragma unroll
    for (int mt = 0; mt < 2; ++mt)
#pragma unroll
      for (int nt = 0; nt < 2; ++nt) {
        int col  = colBase + wn * 32 + nt * 16 + l15;
        float bb = bias[col];
#pragma unroll
        for (int i = 0; i < 8; ++i) {
          float v = c[mt][nt][i] + bb;
          s += v * v;
        }
      }
#pragma unroll
    for (int m = 16; m >= 1; m >>= 1) s += __shfl_xor(s, m, 32);
    if (lane == 0) atomicAdd(accum, s);
  } else {
    float racc[2][8];
#pragma unroll
    for (int mt = 0; mt < 2; ++mt)
#pragma unroll
      for (int i = 0; i < 8; ++i) racc[mt][i] = 0.f;

#pragma unroll
    for (int mt = 0; mt < 2; ++mt)
#pragma unroll
      for (int nt = 0; nt < 2; ++nt) {
        int col  = colBase + wn * 32 + nt * 16 + l15;
        float bb = bias[col];
        float mr = meanref[col];
#pragma unroll
        for (int i = 0; i < 8; ++i) {
          float v = c[mt][nt][i] + bb;
          racc[mt][i] += v * v - 2.0f * v * mr;
        }
      }
    // reduce each row across its 16 lanes (masks < 16 keep half-waves separate)
#pragma unroll
    for (int mt = 0; mt < 2; ++mt)
#pragma unroll
      for (int i = 0; i < 8; ++i) {
        float s = racc[mt][i];
        s += __shfl_xor(s, 1, 32);
        s += __shfl_xor(s, 2, 32);
        s += __shfl_xor(s, 4, 32);
        s += __shfl_xor(s, 8, 32);
        if (l15 == 0) {
          int row = rowBase + wm * 32 + mt * 16 + i + 8 * h;
          atomicAdd(accum + row, s);  // global_atomic_add_f32
        }
      }
  }
}

// ---------------- launch ----------------

extern "C" void kernel_launch(void* const* d_in, const int* in_sizes, int n_in,
                              void* d_out, int out_size, void* d_ws, size_t ws_size,
                              hipStream_t stream) {
  const float* x    = (const float*)d_in[0];  // (4096, 1024)
  const float* refs = (const float*)d_in[1];  // (4096, 1024)
  const float* W    = (const float*)d_in[2];  // (1024, 1024)
  const float* bias = (const float*)d_in[3];  // (1024,)
  float* out = (float*)d_out;                 // (4096,)
  float* ws  = (float*)d_ws;

  // 0) zero accumulators (S, mean_ref, sumsq)
  zero_ws_kernel<<<dim3((WS_NZERO + 255) / 256), 256, 0, stream>>>(ws);
  // 1) S = colsum(refs), row-partitioned (128 blocks)
  colsum_kernel<<<dim3(DIN / 256, 32), 256, 0, stream>>>(refs, ws);
  // 2) mean_ref = (S/R)@W + b, K-partitioned (32 blocks) + finalize
  meanref_partial_kernel<<<dim3(DOUT / 256, 8), 256, 0, stream>>>(W, ws);
  meanref_finalize_kernel<<<dim3(DOUT / 256), 256, 0, stream>>>(bias, ws);
  // 3) sumsq over vrefs = refs@W + b (WMMA, scalar reduction)
  dim3 grid(NROW / BM, DOUT / BN);   // 32 x 16 = 512 blocks
  wmma_gemm_kernel<0><<<grid, 256, 0, stream>>>(refs, W, bias, nullptr,
                                                ws + WS_SUMSQ);
  // 4) out[b] = mean_sq_ref = sumsq / R
  init_out_kernel<<<dim3(NROW / 256), 256, 0, stream>>>(ws, out);
  // 5) out[b] += sum_d vin^2 - 2*vin*mean_ref  (WMMA, per-row reduction)
  wmma_gemm_kernel<1><<<grid, 256, 0, stream>>>(x, W, bias, ws + WS_MEANREF,
                                                out);
}